// GCN_33552284516378
// MI455X (gfx1250) — compile-verified
//
#include <hip/hip_runtime.h>

// GCN forward on MI455X (gfx1250, wave32).
//
// Rewrite: segsum(h[src]) @ W == segsum((h @ W)[src])  -> project first, then
// aggregate 64-wide rows. Everything is L2-resident (192MB L2); the edge
// scatter dominates (~1.2GB L2 traffic). Dense projections use
// V_WMMA_F32_16X16X4_F32 (exact f32 math, matches fp32 reference).

typedef __attribute__((ext_vector_type(2))) float v2f;
typedef __attribute__((ext_vector_type(8))) float v8f;

// ---------------------------------------------------------------- utilities
__global__ void zero_f32(float* __restrict__ p, int n) {
  int i = blockIdx.x * blockDim.x + threadIdx.x;
  if (i < n) p[i] = 0.0f;
}

__global__ void degree_k(const int* __restrict__ src, const int* __restrict__ dst,
                         float* __restrict__ dout, float* __restrict__ din, int nE) {
  int e = blockIdx.x * blockDim.x + threadIdx.x;
  if (e >= nE) return;
  atomicAdd(&dout[src[e]], 1.0f);
  atomicAdd(&din[dst[e]], 1.0f);
}

__global__ void norm_k(float* __restrict__ d, int n) {
  int i = blockIdx.x * blockDim.x + threadIdx.x;
  if (i < n) d[i] = rsqrtf(fmaxf(d[i], 1.0f));
}

// --------------------------------------------------- WMMA GEMM: P = A @ W
// A: M x K (row major), W: K x 64 (row major), P: M x 64. K % 4 == 0, M % 16 == 0.
// One wave computes a 16(M) x 64(N) strip with 4 accumulators.
// Fragment layouts per CDNA5 ISA 7.12.2:
//  A (16x4 f32, 2 VGPR): lanes 0-15 -> M=lane, K={k,k+1}; lanes 16-31 -> K={k+2,k+3}
//  B (4x16 f32, 2 VGPR): VGPRv holds row K = v + (lane>=16 ? 2 : 0), col N = lane&15
//  C/D (16x16 f32, 8 VGPR): VGPRv -> row = v + (lane>=16 ? 8 : 0), col = lane&15
__global__ void gemm_wmma_f32(const float* __restrict__ A, const float* __restrict__ W,
                              float* __restrict__ P, int M, int K) {
  int wave = blockIdx.x * (blockDim.x >> 5) + (threadIdx.x >> 5);
  int lane = threadIdx.x & 31;
  int row0 = wave << 4;
  if (row0 >= M) return;                 // wave-uniform: EXEC stays all-ones
  int ml = lane & 15;
  int hi = lane >> 4;                    // 0 or 1
  int kh = hi * 2;

  v8f c0 = {}, c1 = {}, c2 = {}, c3 = {};
  const float* arow = A + (size_t)(row0 + ml) * K + kh;

  for (int k = 0; k < K; k += 4) {
    v2f a; a.x = arow[k]; a.y = arow[k + 1];
    const float* w0 = W + (size_t)(k + kh) * 64 + ml;   // row k+kh
    const float* w1 = w0 + 64;                          // row k+kh+1
    v2f b0, b1, b2, b3;
    b0.x = w0[0];  b0.y = w1[0];
    b1.x = w0[16]; b1.y = w1[16];
    b2.x = w0[32]; b2.y = w1[32];
    b3.x = w0[48]; b3.y = w1[48];
    c0 = __builtin_amdgcn_wmma_f32_16x16x4_f32(false, a, false, b0, (short)0, c0, false, false);
    c1 = __builtin_amdgcn_wmma_f32_16x16x4_f32(false, a, false, b1, (short)0, c1, false, false);
    c2 = __builtin_amdgcn_wmma_f32_16x16x4_f32(false, a, false, b2, (short)0, c2, false, false);
    c3 = __builtin_amdgcn_wmma_f32_16x16x4_f32(false, a, false, b3, (short)0, c3, false, false);
  }

  int rbase = row0 + hi * 8;
  for (int v = 0; v < 8; ++v) {
    float* orow = P + (size_t)(rbase + v) * 64 + ml;
    orow[0]  = c0[v];
    orow[16] = c1[v];
    orow[32] = c2[v];
    orow[48] = c3[v];
  }
}

// ------------------------------------------- edge scatter: agg[dst] += onorm[src]*P[src]
// 16 lanes per edge, each lane moves one float4 (64 floats per edge).
__global__ void scatter_k(const float* __restrict__ P, const int* __restrict__ src,
                          const int* __restrict__ dst, const float* __restrict__ onorm,
                          float* __restrict__ agg, int nE) {
  int t = blockIdx.x * blockDim.x + threadIdx.x;
  int e = t >> 4;
  if (e >= nE) return;
  int f0 = (t & 15) * 4;
  int s = src[e], d = dst[e];
  float sc = onorm[s];
  const float4 v = *(const float4*)(P + (size_t)s * 64 + f0);
  float* ag = agg + (size_t)d * 64 + f0;
  atomicAdd(ag + 0, v.x * sc);
  atomicAdd(ag + 1, v.y * sc);
  atomicAdd(ag + 2, v.z * sc);
  atomicAdd(ag + 3, v.w * sc);
}

// ----------------------------------------- h = relu(agg * in_norm[row] + bias[col])
__global__ void relu_norm_bias_k(const float* __restrict__ agg, const float* __restrict__ inorm,
                                 const float* __restrict__ bias, float* __restrict__ h, int total) {
  int t = blockIdx.x * blockDim.x + threadIdx.x;
  if (t >= total) return;
  int n = t >> 6, f = t & 63;
  float v = agg[t] * inorm[n] + bias[f];
  h[t] = v > 0.0f ? v : 0.0f;
}

// ------------------------------------------- classifier: out = h.reshape(B,D) @ Wc + bc
__global__ void classifier_k(const float* __restrict__ h, const float* __restrict__ Wc,
                             const float* __restrict__ bc, float* __restrict__ out, int D) {
  int b = blockIdx.x;
  int t = threadIdx.x;
  const float* row = h + (size_t)b * D;
  float s0 = 0.0f, s1 = 0.0f;
  for (int i = t; i < D; i += blockDim.x) {
    float v = row[i];
    s0 += v * Wc[2 * i + 0];
    s1 += v * Wc[2 * i + 1];
  }
  __shared__ float r0[256], r1[256];
  r0[t] = s0; r1[t] = s1;
  __syncthreads();
  for (int off = 128; off > 0; off >>= 1) {
    if (t < off) { r0[t] += r0[t + off]; r1[t] += r1[t + off]; }
    __syncthreads();
  }
  if (t == 0) {
    out[2 * b + 0] = r0[0] + bc[0];
    out[2 * b + 1] = r1[0] + bc[1];
  }
}

// ---------------------------------------------------------------- launcher
extern "C" void kernel_launch(void* const* d_in, const int* in_sizes, int n_in,
                              void* d_out, int out_size, void* d_ws, size_t ws_size,
                              hipStream_t stream) {
  const float* feat = (const float*)d_in[0];   // (N, 116)
  const int*   src  = (const int*)  d_in[1];   // (E,)
  const int*   dst  = (const int*)  d_in[2];   // (E,)
  const float* W1   = (const float*)d_in[4];   // (116, 64)
  const float* b1   = (const float*)d_in[5];   // (64,)
  const float* W2   = (const float*)d_in[6];   // (64, 64)
  const float* b2   = (const float*)d_in[7];   // (64,)
  const float* Wc   = (const float*)d_in[8];   // (7424, 2)
  const float* bc   = (const float*)d_in[9];   // (2,)
  float* out = (float*)d_out;

  const int K1 = 116;
  const int nN = in_sizes[0] / K1;             // 59392
  const int nE = in_sizes[1];                  // 1187840
  const int B  = out_size / 2;                 // 512
  const int D  = (nN / B) * 64;                // 7424

  // workspace layout (floats): onorm[N] | inorm[N] | P[N*64] | agg[N*64] | h[N*64]
  float* ws    = (float*)d_ws;
  float* onorm = ws;
  float* inorm = ws + nN;
  float* P     = ws + 2 * nN;
  float* agg   = P + (size_t)nN * 64;
  float* h     = agg + (size_t)nN * 64;

  const int TB = 256;
  int nF = nN * 64;
  dim3 blkF((nF + TB - 1) / TB);
  dim3 blkN2((2 * nN + TB - 1) / TB);
  dim3 blkE((nE + TB - 1) / TB);
  dim3 blkS(((nE * 16) + TB - 1) / TB);
  dim3 gemmGrid((nN / 16 + 7) / 8);            // 8 waves (16 rows each) per block

  // degrees -> norms
  zero_f32<<<blkN2, TB, 0, stream>>>(onorm, 2 * nN);
  degree_k<<<blkE, TB, 0, stream>>>(src, dst, onorm, inorm, nE);
  norm_k<<<blkN2, TB, 0, stream>>>(onorm, 2 * nN);

  // layer 1: P = feat @ W1; agg = segsum((onorm*P)[src] -> dst); h = relu(inorm*agg + b1)
  gemm_wmma_f32<<<gemmGrid, TB, 0, stream>>>(feat, W1, P, nN, K1);
  zero_f32<<<blkF, TB, 0, stream>>>(agg, nF);
  scatter_k<<<blkS, TB, 0, stream>>>(P, src, dst, onorm, agg, nE);
  relu_norm_bias_k<<<blkF, TB, 0, stream>>>(agg, inorm, b1, h, nF);

  // layer 2
  gemm_wmma_f32<<<gemmGrid, TB, 0, stream>>>(h, W2, P, nN, 64);
  zero_f32<<<blkF, TB, 0, stream>>>(agg, nF);
  scatter_k<<<blkS, TB, 0, stream>>>(P, src, dst, onorm, agg, nE);
  relu_norm_bias_k<<<blkF, TB, 0, stream>>>(agg, inorm, b2, h, nF);

  // classifier
  classifier_k<<<dim3(B), TB, 0, stream>>>(h, Wc, bc, out, D);
}